// adapt_layer_off_39943195853000
// MI455X (gfx1250) — compile-verified
//
#include <hip/hip_runtime.h>
#include <hip/hip_bf16.h>
#include <math.h>

typedef __attribute__((ext_vector_type(2))) float v2f;
typedef __attribute__((ext_vector_type(8))) float v8f;

#define B_   32
#define C_   64
#define N_   8192
#define S_   64      // NUM_NODE
#define K_   64      // NSAMPLE
#define R2_  0.09f   // RADIUS^2
#define BN_EPS_ 1e-5f

// d_out layout (floats): [output_fea B*128*N][node_fea B*64*64][node_offset B*3*64]
#define OUT_FEA_     0
#define OUT_NODEFEA_ (B_*128*N_)
#define OUT_NODEOFF_ (B_*128*N_ + B_*64*64)

// ---------------------------------------------------------------------------
// 1. Farthest point sampling: one block per batch.
// ---------------------------------------------------------------------------
__global__ void k_fps(const float* __restrict__ loc, int* __restrict__ fidx) {
  const int b = blockIdx.x;
  const int tid = threadIdx.x;                    // 256 threads
  __shared__ float dist[N_];
  __shared__ float rmax[256];
  __shared__ int   ridx[256];
  __shared__ float cent[3];
  __shared__ int   s_far;
  const float* lx = loc + (size_t)b * 3 * N_;
  for (int i = tid; i < N_; i += 256) dist[i] = 1e10f;
  if (tid == 0) s_far = 0;
  __syncthreads();
  for (int it = 0; it < S_; ++it) {
    int far = s_far;
    if (tid == 0) fidx[b * S_ + it] = far;
    if (tid < 3) cent[tid] = lx[tid * N_ + far];
    __syncthreads();
    const float cx = cent[0], cy = cent[1], cz = cent[2];
    float bm = -1.0f; int bi = 0;
    for (int i = 0; i < N_ / 256; ++i) {
      const int p = i * 256 + tid;
      const float dx = lx[p] - cx, dy = lx[N_ + p] - cy, dz = lx[2 * N_ + p] - cz;
      const float nd = fminf(dist[p], dx * dx + dy * dy + dz * dz);
      dist[p] = nd;
      if (nd > bm) { bm = nd; bi = p; }           // ascending p -> first max kept
    }
    rmax[tid] = bm; ridx[tid] = bi;
    __syncthreads();
    for (int s = 128; s > 0; s >>= 1) {
      if (tid < s) {
        const float om = rmax[tid + s]; const int oi = ridx[tid + s];
        if (om > rmax[tid] || (om == rmax[tid] && oi < ridx[tid])) { rmax[tid] = om; ridx[tid] = oi; }
      }
      __syncthreads();
    }
    if (tid == 0) s_far = ridx[0];
    __syncthreads();
  }
}

// ---------------------------------------------------------------------------
// 2. Gather node locations + node features.
// ---------------------------------------------------------------------------
__global__ void k_gather_nodes(const float* __restrict__ loc, const float* __restrict__ fea,
                               const int* __restrict__ fidx,
                               float* __restrict__ floc, float* __restrict__ ffea) {
  const int b = blockIdx.x, s = threadIdx.x;      // 64 threads
  const int idx = fidx[b * S_ + s];
  for (int c = 0; c < 3; ++c)
    floc[(b * 3 + c) * S_ + s] = loc[((size_t)b * 3 + c) * N_ + idx];
  for (int c = 0; c < C_; ++c)
    ffea[(b * C_ + c) * S_ + s] = fea[((size_t)b * C_ + c) * N_ + idx];
}

// ---------------------------------------------------------------------------
// 3. Ball query (radius): first 64 indices (ascending) inside radius.
// ---------------------------------------------------------------------------
__global__ void k_ball(const float* __restrict__ loc, const float* __restrict__ floc,
                       int* __restrict__ gidx) {
  const int bs = blockIdx.x, b = bs >> 6, s = bs & 63;
  const int tid = threadIdx.x;                    // 256 threads
  __shared__ unsigned int mask[N_ / 32];
  mask[tid] = 0u;
  __syncthreads();
  const float cx = floc[(b * 3 + 0) * S_ + s];
  const float cy = floc[(b * 3 + 1) * S_ + s];
  const float cz = floc[(b * 3 + 2) * S_ + s];
  const float* lx = loc + (size_t)b * 3 * N_;
  for (int i = 0; i < N_ / 256; ++i) {
    const int p = i * 256 + tid;
    const float dx = lx[p] - cx, dy = lx[N_ + p] - cy, dz = lx[2 * N_ + p] - cz;
    if (dx * dx + dy * dy + dz * dz <= R2_)
      atomicOr(&mask[p >> 5], 1u << (p & 31));
  }
  __syncthreads();
  if (tid == 0) {
    int cnt = 0, first = -1;
    for (int w = 0; w < N_ / 32 && cnt < K_; ++w) {
      unsigned int m = mask[w];
      while (m && cnt < K_) {
        const int p = w * 32 + (__ffs(m) - 1);
        if (first < 0) first = p;
        gidx[bs * K_ + cnt] = p;
        ++cnt;
        m &= m - 1u;
      }
    }
    if (first < 0) first = 0;
    for (; cnt < K_; ++cnt) gidx[bs * K_ + cnt] = first;
  }
}

// ---------------------------------------------------------------------------
// 4. Offset MLP (tanh(W_off * gfea)) * gloc, mean -> node_offset, node_loc.
// ---------------------------------------------------------------------------
__global__ void k_offset(const float* __restrict__ loc, const float* __restrict__ fea,
                         const float* __restrict__ Woff, const float* __restrict__ floc,
                         const float* __restrict__ ffea, const int* __restrict__ gidx,
                         float* __restrict__ nloc, float* __restrict__ noff) {
  const int bs = blockIdx.x, b = bs >> 6, s = bs & 63;
  const int k = threadIdx.x;                      // 64 threads, one per sample
  __shared__ float W[3 * C_];
  __shared__ float red[3][64];
  for (int i = k; i < 3 * C_; i += 64) W[i] = Woff[i];
  __syncthreads();
  const int idx = gidx[bs * K_ + k];
  const float* fb  = fea  + (size_t)b * C_ * N_;
  const float* ffb = ffea + (size_t)b * C_ * S_;
  float a0 = 0.f, a1 = 0.f, a2 = 0.f;
  for (int c = 0; c < C_; ++c) {
    const float g = fb[c * N_ + idx] - ffb[c * S_ + s];
    a0 += W[0 * C_ + c] * g;
    a1 += W[1 * C_ + c] * g;
    a2 += W[2 * C_ + c] * g;
  }
  const float t[3] = { tanhf(a0), tanhf(a1), tanhf(a2) };
  for (int o = 0; o < 3; ++o) {
    const float gl = loc[((size_t)b * 3 + o) * N_ + idx] - floc[(b * 3 + o) * S_ + s];
    red[o][k] = t[o] * gl;
  }
  __syncthreads();
  for (int st = 32; st > 0; st >>= 1) {
    if (k < st)
      for (int o = 0; o < 3; ++o) red[o][k] += red[o][k + st];
    __syncthreads();
  }
  if (k < 3) {
    const float off = red[k][0] * (1.0f / 64.0f);
    noff[(b * 3 + k) * S_ + s] = off;
    nloc[(b * 3 + k) * S_ + s] = floc[(b * 3 + k) * S_ + s] + off;
  }
}

// ---------------------------------------------------------------------------
// 5. KNN (k=64 nearest of N=8192) per node: stable argsort top-64.
// ---------------------------------------------------------------------------
__global__ void k_knn(const float* __restrict__ loc, const float* __restrict__ nloc,
                      int* __restrict__ gidx2) {
  const int bs = blockIdx.x, b = bs >> 6, s = bs & 63;
  const int tid = threadIdx.x;                    // 256 threads
  __shared__ float dist[N_];
  __shared__ float rmin[256];
  __shared__ int   rid[256];
  const float cx = nloc[(b * 3 + 0) * S_ + s];
  const float cy = nloc[(b * 3 + 1) * S_ + s];
  const float cz = nloc[(b * 3 + 2) * S_ + s];
  const float* lx = loc + (size_t)b * 3 * N_;
  for (int i = 0; i < N_ / 256; ++i) {
    const int p = i * 256 + tid;
    const float dx = lx[p] - cx, dy = lx[N_ + p] - cy, dz = lx[2 * N_ + p] - cz;
    dist[p] = dx * dx + dy * dy + dz * dz;
  }
  __syncthreads();
  for (int j = 0; j < K_; ++j) {
    float bm = 3.4e38f; int bi = N_;
    for (int i = 0; i < N_ / 256; ++i) {
      const int p = i * 256 + tid;
      const float d = dist[p];
      if (d < bm || (d == bm && p < bi)) { bm = d; bi = p; }
    }
    rmin[tid] = bm; rid[tid] = bi;
    __syncthreads();
    for (int st = 128; st > 0; st >>= 1) {
      if (tid < st) {
        const float om = rmin[tid + st]; const int oi = rid[tid + st];
        if (om < rmin[tid] || (om == rmin[tid] && oi < rid[tid])) { rmin[tid] = om; rid[tid] = oi; }
      }
      __syncthreads();
    }
    if (tid == 0) { gidx2[bs * K_ + j] = rid[0]; dist[rid[0]] = 3.3e38f; }
    __syncthreads();
  }
}

// ---------------------------------------------------------------------------
// 6a. Full GEMM y = W_res * x + b via V_WMMA_F32_16X16X4_F32, accumulating
//     deterministic per-channel sum / sumsq partials (y never stored).
//     8 waves per block; each wave computes a 64x16 (channels x columns) tile.
// ---------------------------------------------------------------------------
__global__ void k_gemm_stats(const float* __restrict__ fea, const float* __restrict__ Wres,
                             const float* __restrict__ bres,
                             float* __restrict__ psum, float* __restrict__ psq) {
  const int tid = threadIdx.x;                    // 256 threads = 8 waves
  const int lane = tid & 31, wv = tid >> 5;
  const int half = lane >> 4, l16 = lane & 15;
  __shared__ float Wl[64 * 65];
  __shared__ float bl[64];
  __shared__ float wsum[8 * 64];
  __shared__ float wsq[8 * 64];
  for (int i = tid; i < 64 * 64; i += 256) Wl[(i >> 6) * 65 + (i & 63)] = Wres[i];
  if (tid < 64) bl[tid] = bres[tid];
  __syncthreads();

  const int col = blockIdx.x * 128 + wv * 16;     // global column (b*N + n)
  const int b = col >> 13, n0 = col & (N_ - 1);
  const float* xb = fea + (size_t)b * C_ * N_;

  v8f acc[4] = {};
  for (int k = 0; k < C_; k += 4) {
    const int kk = k + half * 2;                  // B matrix: 4x16 f32 layout
    v2f bm;
    bm.x = xb[(kk + 0) * N_ + n0 + l16];
    bm.y = xb[(kk + 1) * N_ + n0 + l16];
#pragma unroll
    for (int mt = 0; mt < 4; ++mt) {
      v2f a;                                      // A matrix: 16x4 f32 layout
      a.x = Wl[(mt * 16 + l16) * 65 + kk + 0];
      a.y = Wl[(mt * 16 + l16) * 65 + kk + 1];
      acc[mt] = __builtin_amdgcn_wmma_f32_16x16x4_f32(
          false, a, false, bm, (short)0, acc[mt], false, false);
    }
  }

  // Per-wave channel sums over this wave's 16 columns (deterministic).
#pragma unroll
  for (int mt = 0; mt < 4; ++mt) {
#pragma unroll
    for (int r = 0; r < 8; ++r) {
      const int m = mt * 16 + r + half * 8;       // C/D layout: M = r + 8*half
      const float val = acc[mt][r] + bl[m];
      float sv = val, sq = val * val;
      for (int d = 8; d >= 1; d >>= 1) {          // reduce within 16-lane half
        sv += __shfl_xor(sv, d, 32);
        sq += __shfl_xor(sq, d, 32);
      }
      if (l16 == 0) { wsum[wv * 64 + m] = sv; wsq[wv * 64 + m] = sq; }
    }
  }
  __syncthreads();
  if (tid < 64) {
    float s = 0.f, q = 0.f;
    for (int w = 0; w < 8; ++w) { s += wsum[w * 64 + tid]; q += wsq[w * 64 + tid]; }
    psum[blockIdx.x * 64 + tid] = s;
    psq[blockIdx.x * 64 + tid] = q;
  }
}

// ---------------------------------------------------------------------------
// 6b. Reduce partials -> per-channel BN scale/shift (deterministic serial).
// ---------------------------------------------------------------------------
__global__ void k_finalize(const float* __restrict__ psum, const float* __restrict__ psq,
                           const float* __restrict__ gamma, const float* __restrict__ beta,
                           float* __restrict__ scale, float* __restrict__ shift) {
  const int o = threadIdx.x;                      // 64 threads
  float s = 0.f, q = 0.f;
  for (int blk = 0; blk < 2048; ++blk) { s += psum[blk * 64 + o]; q += psq[blk * 64 + o]; }
  const float inv = 1.0f / (float)(B_ * N_);
  const float mean = s * inv;
  const float var = q * inv - mean * mean;
  const float sc = gamma[o] * rsqrtf(var + BN_EPS_);
  scale[o] = sc;
  shift[o] = beta[o] - mean * sc;
}

// ---------------------------------------------------------------------------
// 7. Per-node gathered GEMM (WMMA) + BN + ReLU + max over samples -> node_fea.
//    512 threads = 16 waves; wave (mt, nt) computes one 16x16 tile of y.
// ---------------------------------------------------------------------------
__global__ void k_node_fea(const float* __restrict__ fea, const float* __restrict__ Wres,
                           const float* __restrict__ bres, const float* __restrict__ scale,
                           const float* __restrict__ shift, const int* __restrict__ gidx2,
                           float* __restrict__ node_fea_out) {
  const int bs = blockIdx.x, b = bs >> 6, s = bs & 63;
  const int tid = threadIdx.x;                    // 512 threads
  const int lane = tid & 31, wv = tid >> 5;
  const int half = lane >> 4, l16 = lane & 15;
  const int mt = wv & 3, nt = wv >> 2;
  __shared__ float Wl[64 * 65];
  __shared__ float Xg[64 * 65];
  __shared__ int   sidx[64];
  __shared__ float bl[64], scl[64], shl[64];
  __shared__ unsigned int maxu[64];
  for (int i = tid; i < 64 * 64; i += 512) Wl[(i >> 6) * 65 + (i & 63)] = Wres[i];
  if (tid < 64) {
    sidx[tid] = gidx2[bs * K_ + tid];
    bl[tid] = bres[tid]; scl[tid] = scale[tid]; shl[tid] = shift[tid];
    maxu[tid] = 0u;                               // ReLU output >= 0
  }
  __syncthreads();
  const float* fb = fea + (size_t)b * C_ * N_;
  for (int i = tid; i < 64 * 64; i += 512) {
    const int c = i >> 6, k = i & 63;
    Xg[c * 65 + k] = fb[c * N_ + sidx[k]];
  }
  __syncthreads();

  v8f acc = {};
  for (int c = 0; c < C_; c += 4) {
    const int cc = c + half * 2;
    v2f a, bm;
    a.x  = Wl[(mt * 16 + l16) * 65 + cc + 0];
    a.y  = Wl[(mt * 16 + l16) * 65 + cc + 1];
    bm.x = Xg[(cc + 0) * 65 + nt * 16 + l16];
    bm.y = Xg[(cc + 1) * 65 + nt * 16 + l16];
    acc = __builtin_amdgcn_wmma_f32_16x16x4_f32(
        false, a, false, bm, (short)0, acc, false, false);
  }

#pragma unroll
  for (int r = 0; r < 8; ++r) {
    const int m = mt * 16 + r + half * 8;
    float val = (acc[r] + bl[m]) * scl[m] + shl[m];
    val = fmaxf(val, 0.0f);                       // ReLU
    for (int d = 8; d >= 1; d >>= 1)              // max over 16 samples in half
      val = fmaxf(val, __shfl_xor(val, d, 32));
    if (l16 == 0) atomicMax(&maxu[m], __float_as_uint(val));  // val >= 0
  }
  __syncthreads();
  if (tid < 64)
    node_fea_out[b * 64 * 64 + tid * 64 + s] = __uint_as_float(maxu[tid]);
}

// ---------------------------------------------------------------------------
// 8. Upsample: 3-NN inverse-distance interp of node_fea; concat [fea, interp].
// ---------------------------------------------------------------------------
__global__ void k_upsample(const float* __restrict__ loc, const float* __restrict__ fea,
                           const float* __restrict__ nloc, const float* __restrict__ nfea,
                           float* __restrict__ outfea) {
  const int b = blockIdx.x >> 5;                  // 32 blocks per batch
  const int nblk = blockIdx.x & 31;
  const int tid = threadIdx.x;                    // 256 threads
  const int n = nblk * 256 + tid;
  __shared__ float nl[3][64];
  __shared__ float nf[64 * 64];
  if (tid < 192) nl[tid / 64][tid % 64] = nloc[b * 3 * 64 + tid];
  for (int i = tid; i < 64 * 64; i += 256) nf[i] = nfea[b * 64 * 64 + i];
  __syncthreads();
  const float px = loc[((size_t)b * 3 + 0) * N_ + n];
  const float py = loc[((size_t)b * 3 + 1) * N_ + n];
  const float pz = loc[((size_t)b * 3 + 2) * N_ + n];
  float d0 = 3.4e38f, d1 = 3.4e38f, d2 = 3.4e38f;
  int i0 = 0, i1 = 0, i2 = 0;
  for (int j = 0; j < S_; ++j) {
    const float dx = px - nl[0][j], dy = py - nl[1][j], dz = pz - nl[2][j];
    const float d = dx * dx + dy * dy + dz * dz;
    if (d < d0)      { d2 = d1; i2 = i1; d1 = d0; i1 = i0; d0 = d; i0 = j; }
    else if (d < d1) { d2 = d1; i2 = i1; d1 = d; i1 = j; }
    else if (d < d2) { d2 = d; i2 = j; }
  }
  float w0 = 1.0f / fmaxf(d0, 1e-10f);
  float w1 = 1.0f / fmaxf(d1, 1e-10f);
  float w2 = 1.0f / fmaxf(d2, 1e-10f);
  const float wi = 1.0f / (w0 + w1 + w2);
  w0 *= wi; w1 *= wi; w2 *= wi;
  float* ob = outfea + (size_t)b * 128 * N_;
  const float* fb = fea + (size_t)b * C_ * N_;
  for (int c = 0; c < C_; ++c) {
    ob[c * N_ + n] = fb[c * N_ + n];
    ob[(64 + c) * N_ + n] =
        w0 * nf[c * 64 + i0] + w1 * nf[c * 64 + i1] + w2 * nf[c * 64 + i2];
  }
}

// ---------------------------------------------------------------------------
// Launch
// ---------------------------------------------------------------------------
extern "C" void kernel_launch(void* const* d_in, const int* in_sizes, int n_in,
                              void* d_out, int out_size, void* d_ws, size_t ws_size,
                              hipStream_t stream) {
  const float* fea   = (const float*)d_in[0];   // (B,64,N,1)
  const float* loc   = (const float*)d_in[1];   // (B,3,N)
  const float* Woff  = (const float*)d_in[2];   // (3,64)
  const float* Wres  = (const float*)d_in[3];   // (64,64)
  const float* bres  = (const float*)d_in[4];   // (64,)
  const float* gamma = (const float*)d_in[5];   // (64,)
  const float* beta  = (const float*)d_in[6];   // (64,)
  float* out = (float*)d_out;

  char* w = (char*)d_ws;
  int*   fidx  = (int*)  (w + 0);               //  2048 ints
  float* floc  = (float*)(w + 8192);            //  6144 f
  float* ffea  = (float*)(w + 32768);           //  131072 f
  int*   gidx  = (int*)  (w + 557056);          //  131072 ints
  float* nloc  = (float*)(w + 1081344);         //  6144 f
  int*   gidx2 = (int*)  (w + 1105920);         //  131072 ints
  float* scale = (float*)(w + 1630208);         //  64 f
  float* shift = (float*)(w + 1630208 + 256);   //  64 f
  float* psum  = (float*)(w + 1630720);         //  2048*64 f
  float* psq   = (float*)(w + 1630720 + 2048 * 64 * 4);

  k_fps<<<B_, 256, 0, stream>>>(loc, fidx);
  k_gather_nodes<<<B_, 64, 0, stream>>>(loc, fea, fidx, floc, ffea);
  k_ball<<<B_ * S_, 256, 0, stream>>>(loc, floc, gidx);
  k_offset<<<B_ * S_, 64, 0, stream>>>(loc, fea, Woff, floc, ffea, gidx,
                                       nloc, out + OUT_NODEOFF_);
  k_knn<<<B_ * S_, 256, 0, stream>>>(loc, nloc, gidx2);
  k_gemm_stats<<<(B_ * N_) / 128, 256, 0, stream>>>(fea, Wres, bres, psum, psq);
  k_finalize<<<1, 64, 0, stream>>>(psum, psq, gamma, beta, scale, shift);
  k_node_fea<<<B_ * S_, 512, 0, stream>>>(fea, Wres, bres, scale, shift, gidx2,
                                          out + OUT_NODEFEA_);
  k_upsample<<<B_ * (N_ / 256), 256, 0, stream>>>(loc, fea, nloc,
                                                  out + OUT_NODEFEA_, out + OUT_FEA_);
}